// SoftBCELoss_57501022158919
// MI455X (gfx1250) — compile-verified
//
#include <hip/hip_runtime.h>
#include <cstdint>

// Problem constants (from reference: B=256, C=2, T=60000)
#define T_LEN   60000
#define B_SZ    256
#define TILE    4096
#define BLOCK   256
#define TILES_PER_ROW ((T_LEN + TILE - 1) / TILE)   // 15

// Collapsed 7-tap circular stencil weights:
// (1+z^2/4)(1+z/2)(1+z^-1/2)(1+z^-2/4) -> [1/8, 5/16, 21/32, 85/64, 21/32, 5/16, 1/8]
#define W0 1.328125f   // 85/64
#define W1 0.65625f    // 21/32
#define W2 0.3125f     // 5/16
#define W3 0.125f      // 1/8

#define LN2F        0.69314718055994531f
#define CLAMP_LOG2  (-144.26950408889634f)   // -100 / ln2

// Fused stencil + BCE in log2 domain. FULL => cnt == TILE (no guards, full unroll).
template <bool FULL>
__device__ __forceinline__ float tile_accum(const float* __restrict__ smem,
                                            const float* __restrict__ prow,
                                            int start, int cnt, int tid)
{
    float acc = 0.0f;
#pragma unroll
    for (int k = 0; k < TILE / BLOCK; ++k) {
        const int j = tid + k * BLOCK;          // strided: conflict-free LDS, coalesced global
        if (FULL || j < cnt) {
            const float* s = &smem[4 + j];
            float tv = W3 * (s[-3] + s[3]);
            tv = fmaf(W2, s[-2] + s[2], tv);
            tv = fmaf(W1, s[-1] + s[1], tv);
            tv = fmaf(W0, s[0], tv);
            tv = fminf(tv, 1.0f);

            const float p   = prow[start + j];
            const float lp2 = fmaxf(__log2f(p),        CLAMP_LOG2);
            const float lq2 = fmaxf(__log2f(1.0f - p), CLAMP_LOG2);
            // bce/ln2 = -(lq2 + t*(lp2-lq2)); scale by ln2 once per thread later
            acc -= fmaf(tv, lp2 - lq2, lq2);
        }
    }
    return acc;
}

__global__ __launch_bounds__(BLOCK)
void bce_tile_kernel(const float* __restrict__ input,
                     const float* __restrict__ target,
                     float* __restrict__ partials)
{
    // Layout: smem[0] pad, smem[1..3] left halo (j=-3..-1),
    //         smem[4 .. 4+cnt) bulk tile (16B-aligned start),
    //         smem[4+cnt .. 4+cnt+2] right halo (j=cnt..cnt+2)
    __shared__ __align__(16) float smem[TILE + 8];

    const int tid  = threadIdx.x;
    const int tile = blockIdx.x;
    const int b    = blockIdx.y;
    const int c    = blockIdx.z;

    const long long row = (long long)(b * 2 + c);
    const float* __restrict__ trow = target + row * T_LEN;
    const float* __restrict__ prow = input  + row * T_LEN;

    const int start = tile * TILE;
    const int cnt   = min(TILE, T_LEN - start);   // T_LEN % 4 == 0 so cnt % 4 == 0

    // ---- Stage bulk tile into LDS via CDNA5 async global->LDS (ASYNCcnt path) ----
    {
        const float* gbase = trow + start;                   // uniform -> SGPR pair
        const int nf4 = cnt >> 2;                            // float4 count
        const uint32_t lds0 = (uint32_t)(uintptr_t)&smem[4]; // low 32 bits = LDS offset
#pragma unroll
        for (int k = 0; k < TILE / (BLOCK * 4); ++k) {
            const int idx4 = tid + k * BLOCK;
            if (idx4 < nf4) {
                const uint32_t boff = (uint32_t)idx4 * 16u;
                asm volatile("global_load_async_to_lds_b128 %0, %1, %2"
                             :
                             : "v"(lds0 + boff), "v"(boff), "s"(gbase)
                             : "memory");
            }
        }
    }

    // ---- Halo (circular wrap at row boundaries), plain LDS stores ----
    if (tid < 3) {
        int g = start - 3 + tid; if (g < 0) g += T_LEN;
        smem[1 + tid] = trow[g];
    } else if (tid < 6) {
        int j = cnt + (tid - 3);
        int g = start + j; if (g >= T_LEN) g -= T_LEN;
        smem[4 + cnt + (tid - 3)] = trow[g];
    }

    asm volatile("s_wait_asynccnt 0x0" ::: "memory");
    __syncthreads();

    // ---- Compute: full-tile fast path (14/15 tiles) vs guarded tail ----
    float acc;
    if (cnt == TILE) acc = tile_accum<true >(smem, prow, start, cnt, tid);
    else             acc = tile_accum<false>(smem, prow, start, cnt, tid);
    acc *= LN2F;   // back to natural-log units (one mul per thread, not per element)

    // ---- Deterministic block reduction (reuse LDS after barrier) ----
    __syncthreads();
    smem[tid] = acc;
    __syncthreads();
#pragma unroll
    for (int s2 = BLOCK / 2; s2 > 0; s2 >>= 1) {
        if (tid < s2) smem[tid] += smem[tid + s2];
        __syncthreads();
    }
    if (tid == 0)
        partials[(c * B_SZ + b) * TILES_PER_ROW + tile] = smem[0];
}

__global__ __launch_bounds__(256)
void finalize_kernel(const float* __restrict__ partials, float* __restrict__ out)
{
    __shared__ float red[256];
    const int tid = threadIdx.x;
    const int per_ch = B_SZ * TILES_PER_ROW;   // 3840 partials per channel
    float sums[2];
#pragma unroll
    for (int c = 0; c < 2; ++c) {
        float a = 0.0f;
        for (int i = tid; i < per_ch; i += 256) a += partials[c * per_ch + i];
        red[tid] = a;
        __syncthreads();
#pragma unroll
        for (int s = 128; s > 0; s >>= 1) {
            if (tid < s) red[tid] += red[tid + s];
            __syncthreads();
        }
        sums[c] = red[0];
        __syncthreads();
    }
    if (tid == 0) {
        const float inv = 0.5f / (float)(B_SZ * T_LEN);  // alpha = gamma = 0.5
        const float tb = sums[0] * inv;
        const float td = sums[1] * inv;
        out[0] = tb + td;   // total
        out[1] = tb;        // total_beat
        out[2] = td;        // total_down
    }
}

extern "C" void kernel_launch(void* const* d_in, const int* in_sizes, int n_in,
                              void* d_out, int out_size, void* d_ws, size_t ws_size,
                              hipStream_t stream)
{
    const float* input  = (const float*)d_in[0];   // (256, 2, 60000) f32
    const float* target = (const float*)d_in[1];   // (256, 2, 60000) f32
    float* partials = (float*)d_ws;                // 2*3840 floats = 30 KB

    dim3 grid(TILES_PER_ROW, B_SZ, 2);             // 7680 blocks
    bce_tile_kernel<<<grid, BLOCK, 0, stream>>>(input, target, partials);
    finalize_kernel<<<1, 256, 0, stream>>>(partials, (float*)d_out);
}